// MultiHeadAttention_7799660609796
// MI455X (gfx1250) — compile-verified
//
#include <hip/hip_runtime.h>
#include <hip/hip_bf16.h>

// MHA forward, fp32, CDNA5 (gfx1250, wave32).
// Shapes: query/key/value (512,16,1024), W* (1024,1024), heads=16, head_dim=64.
// d_out = attn_output (512*16*1024 floats) ++ attn_weights (16*16*512*512 floats).
//
// Structure (roofline-driven):
//  - 4x WMMA GEMM (X @ W.T + b) with async-to-LDS double-buffered tiles
//  - 1x fused attention: scores -> mask/scale -> softmax -> AV, score band in VGPRs,
//    attn_weights written exactly once; K/V chunks streamed via async-to-LDS.

#define NQ     512
#define NKV    512
#define BATCH  16
#define EMBED  1024
#define HEADS  16
#define HDIM   64
#define TOK    (NQ * BATCH)        // 8192 token rows
#define ROWSTRIDE (BATCH * EMBED)  // 16384 floats between sequence positions

typedef __attribute__((ext_vector_type(2))) float v2f;
typedef __attribute__((ext_vector_type(8))) float v8f;

// D = A(16x4 f32) * B(4x16 f32) + C  -> v_wmma_f32_16x16x4_f32
__device__ __forceinline__ v8f wmma4(v2f a, v2f b, v8f c) {
    return __builtin_amdgcn_wmma_f32_16x16x4_f32(
        /*neg_a=*/false, a, /*neg_b=*/false, b,
        /*c_mod=*/(short)0, c, /*reuse_a=*/false, /*reuse_b=*/false);
}

// Async copy 16B global -> LDS (GLOBAL_LOAD_ASYNC_TO_LDS_B128, tracked by ASYNCcnt).
// ldsOff: byte offset within the workgroup LDS allocation (low 32 bits of the
// flat address of a __shared__ object are exactly this offset).
__device__ __forceinline__ void async_ld16(const float* gptr, void* lptr) {
    const unsigned lds = (unsigned)(unsigned long long)lptr;
    asm volatile("global_load_async_to_lds_b128 %0, %1, off"
                 :: "v"(lds), "v"((unsigned long long)gptr)
                 : "memory");
}
#define WAIT_ASYNC(n) asm volatile("s_wait_asynccnt %0" :: "n"(n) : "memory")

// ---------------------------------------------------------------------------
// Y[m, n] = sum_k X[m, k] * W[n, k] + bias[n]      (torch Linear: X @ W.T + b)
// M = TOK (8192), K = N = EMBED (1024).
// Block: 256 threads = 8 waves; block tile 64(M) x 32(N); K-step 32.
// Async-to-LDS double-buffered pipeline: 3 async instructions per batch per
// wave; in-order completion => s_wait_asynccnt 3 means the older batch landed.
// ---------------------------------------------------------------------------
__global__ __launch_bounds__(256)
void gemm_xwT_bias(const float* __restrict__ X, const float* __restrict__ W,
                   const float* __restrict__ bias, float* __restrict__ Y) {
    __shared__ alignas(16) float As[2][64][36];   // [buf][m][k], +4 pad
    __shared__ alignas(16) float Bs[2][32][36];   // [buf][n][k], +4 pad

    const int tid    = threadIdx.x;
    const int lane   = tid & 31;
    const int wave   = tid >> 5;
    const int lane16 = lane & 15;   // M index (A) and N index (B/C)
    const int half   = lane >> 4;   // K-half (A/B), M-half (C/D)
    const int m0 = (wave & 3) * 16;
    const int n0 = (wave >> 2) * 16;
    const int mBase = blockIdx.y * 64;
    const int nBase = blockIdx.x * 32;

    const int row = tid >> 3;         // 0..31
    const int col = (tid & 7) * 4;    // 0..28

    // prologue: batch for k0 = 0 into buffer 0
    {
        #pragma unroll
        for (int r = 0; r < 2; ++r) {
            const int rr = row + 32 * r;
            async_ld16(&X[(size_t)(mBase + rr) * EMBED + col], &As[0][rr][col]);
        }
        async_ld16(&W[(size_t)(nBase + row) * EMBED + col], &Bs[0][row][col]);
    }

    v8f c = {};
    const int STEPS = EMBED / 32;     // 32

    for (int s = 0; s < STEPS; ++s) {
        const int cur = s & 1;
        __syncthreads();              // readers of buf[1-cur] (step s-1) done
        if (s + 1 < STEPS) {
            const int k0 = (s + 1) * 32;
            #pragma unroll
            for (int r = 0; r < 2; ++r) {
                const int rr = row + 32 * r;
                async_ld16(&X[(size_t)(mBase + rr) * EMBED + k0 + col],
                           &As[1 - cur][rr][col]);
            }
            async_ld16(&W[(size_t)(nBase + row) * EMBED + k0 + col],
                       &Bs[1 - cur][row][col]);
            WAIT_ASYNC(3);            // batch feeding buf[cur] has landed
        } else {
            WAIT_ASYNC(0);
        }
        __syncthreads();              // buf[cur] visible to all waves

        #pragma unroll
        for (int kk = 0; kk < 8; ++kk) {
            const int kb = kk * 4 + 2 * half;              // K = v + 2*(lane/16)
            v2f a = *(const v2f*)&As[cur][m0 + lane16][kb];
            v2f b = *(const v2f*)&Bs[cur][n0 + lane16][kb];
            c = wmma4(a, b, c);
        }
    }

    const int n = nBase + n0 + lane16;
    const float bv = bias[n];
    #pragma unroll
    for (int v = 0; v < 8; ++v) {
        const int m = mBase + m0 + v + 8 * half;           // C/D layout
        Y[(size_t)m * EMBED + n] = c[v] + bv;
    }
}

// ---------------------------------------------------------------------------
// Fused attention for one (b,h) and one 64-row i-band:
//   S = (Q Kh^T)/8 + mask   (64 x 512, kept in VGPRs: 8 chunk-tiles per wave)
//   W = softmax_rows(S)     (lane-shuffle + LDS cross-wave reduction)
//   write W to attn_weights output (single pass)
//   O = W @ Vh              (restage each normalized 64x64 chunk via LDS)
// Block: 512 threads = 16 waves arranged 4(M/i) x 4(N: j-subtile / d-subtile).
// ---------------------------------------------------------------------------
__global__ __launch_bounds__(512)
void attn_fused(const float* __restrict__ Q, const float* __restrict__ K,
                const float* __restrict__ V, const int* __restrict__ kpm,
                float* __restrict__ Wgt, float* __restrict__ Out) {
    __shared__ alignas(16) float Qs[64][68];    // [i][d]
    __shared__ alignas(16) float KVs[64][68];   // [j][d], K then V chunks
    __shared__ alignas(16) float Ws[64][68];    // [i][j] normalized-weight staging
    __shared__ float red[64][4];                // per-row cross-wave reduction

    const int tid    = threadIdx.x;
    const int lane   = tid & 31;
    const int wave   = tid >> 5;
    const int lane16 = lane & 15;
    const int half   = lane >> 4;
    const int wn     = wave >> 2;
    const int m0 = (wave & 3) * 16;
    const int n0 = wn * 16;
    const int i0 = blockIdx.x * 64;
    const int bh = blockIdx.y;
    const int b  = bh >> 4;
    const int h  = bh & 15;
    const size_t cbase = (size_t)b * EMBED + (size_t)h * HDIM;

    const int lrow = tid >> 4;          // 0..31 (tile-load row)
    const int lcol = (tid & 15) * 4;    // 0..60 (tile-load col, 16B)

    // ---- async-load Q band (64 x 64) once; waited before first compute ----
    #pragma unroll
    for (int r = 0; r < 2; ++r) {
        const int rr = lrow + 32 * r;
        async_ld16(&Q[(size_t)(i0 + rr) * ROWSTRIDE + cbase + lcol], &Qs[rr][lcol]);
    }

    // ---- scores: 8 chunks of 64 j-columns, kept in registers ----
    v8f sc[8];
    #pragma unroll
    for (int cch = 0; cch < 8; ++cch) {
        const int j0 = cch * 64;
        __syncthreads();   // prior chunk's KVs readers done
        #pragma unroll
        for (int r = 0; r < 2; ++r) {
            const int rr = lrow + 32 * r;
            async_ld16(&K[(size_t)(j0 + rr) * ROWSTRIDE + cbase + lcol],
                       &KVs[rr][lcol]);
        }
        WAIT_ASYNC(0);     // K chunk (and, on chunk 0, the Q band) landed
        __syncthreads();

        v8f c = {};
        #pragma unroll
        for (int kk = 0; kk < 16; ++kk) {
            const int kb = kk * 4 + 2 * half;
            v2f a  = *(const v2f*)&Qs[m0 + lane16][kb];
            v2f bb = *(const v2f*)&KVs[n0 + lane16][kb];   // B[d][j] = K[j][d]
            c = wmma4(a, bb, c);
        }
        const float maskadd = kpm[b * NKV + j0 + n0 + lane16] ? -1e9f : 0.0f;
        #pragma unroll
        for (int v = 0; v < 8; ++v) sc[cch][v] = c[v] * 0.125f + maskadd;
    }

    // ---- softmax over j (rows of the 64x512 band) ----
    // C-layout: reg v holds row m0+v+8*half, column n0+lane16 (+64*chunk).
    float rowmax[8];
    #pragma unroll
    for (int v = 0; v < 8; ++v) {
        float m = sc[0][v];
        #pragma unroll
        for (int cch = 1; cch < 8; ++cch) m = fmaxf(m, sc[cch][v]);
        #pragma unroll
        for (int off = 1; off < 16; off <<= 1)
            m = fmaxf(m, __shfl_xor(m, off, 32));   // reduce 16 cols within half
        rowmax[v] = m;
    }
    __syncthreads();
    if (lane16 == 0) {
        #pragma unroll
        for (int v = 0; v < 8; ++v) red[m0 + v + 8 * half][wn] = rowmax[v];
    }
    __syncthreads();
    #pragma unroll
    for (int v = 0; v < 8; ++v) {
        const int rw = m0 + v + 8 * half;
        rowmax[v] = fmaxf(fmaxf(red[rw][0], red[rw][1]),
                          fmaxf(red[rw][2], red[rw][3]));
    }
    __syncthreads();   // all rowmax reads done before red is reused for sums

    float rowsum[8];
    #pragma unroll
    for (int v = 0; v < 8; ++v) {
        float s = 0.0f;
        #pragma unroll
        for (int cch = 0; cch < 8; ++cch) {
            const float p = __expf(sc[cch][v] - rowmax[v]);
            sc[cch][v] = p;
            s += p;
        }
        #pragma unroll
        for (int off = 1; off < 16; off <<= 1)
            s += __shfl_xor(s, off, 32);
        rowsum[v] = s;
    }
    if (lane16 == 0) {
        #pragma unroll
        for (int v = 0; v < 8; ++v) red[m0 + v + 8 * half][wn] = rowsum[v];
    }
    __syncthreads();
    #pragma unroll
    for (int v = 0; v < 8; ++v) {
        const int rw = m0 + v + 8 * half;
        const float s = red[rw][0] + red[rw][1] + red[rw][2] + red[rw][3];
        const float inv = 1.0f / s;
        #pragma unroll
        for (int cch = 0; cch < 8; ++cch) sc[cch][v] *= inv;
    }

    // ---- write weights (single pass) + AV accumulation ----
    v8f o = {};
    #pragma unroll
    for (int cch = 0; cch < 8; ++cch) {
        const int j0 = cch * 64;
        __syncthreads();   // prior chunk's Ws/KVs readers done
        #pragma unroll
        for (int v = 0; v < 8; ++v) {
            const int i = i0 + m0 + v + 8 * half;
            const float w = sc[cch][v];
            Ws[m0 + v + 8 * half][n0 + lane16] = w;             // stage [i][j]
            Wgt[((size_t)bh * NQ + i) * NKV + j0 + n0 + lane16] = w;
        }
        #pragma unroll
        for (int r = 0; r < 2; ++r) {
            const int rr = lrow + 32 * r;
            async_ld16(&V[(size_t)(j0 + rr) * ROWSTRIDE + cbase + lcol],
                       &KVs[rr][lcol]);
        }
        WAIT_ASYNC(0);     // V chunk landed (Ws dscnt handled by the barrier)
        __syncthreads();

        #pragma unroll
        for (int kk = 0; kk < 16; ++kk) {
            const int kb = kk * 4 + 2 * half;
            v2f a = *(const v2f*)&Ws[m0 + lane16][kb];          // A[i][j]
            v2f bb;
            bb.x = KVs[kb + 0][n0 + lane16];                    // B[j][d] = V[j][d]
            bb.y = KVs[kb + 1][n0 + lane16];
            o = wmma4(a, bb, o);
        }
    }

    // ---- write attention result in (n, N, E) token layout ----
    const int d = n0 + lane16;
    #pragma unroll
    for (int v = 0; v < 8; ++v) {
        const int i = i0 + m0 + v + 8 * half;
        Out[(size_t)(i * BATCH + b) * EMBED + (size_t)h * HDIM + d] = o[v];
    }
}

// ---------------------------------------------------------------------------
extern "C" void kernel_launch(void* const* d_in, const int* in_sizes, int n_in,
                              void* d_out, int out_size, void* d_ws, size_t ws_size,
                              hipStream_t stream) {
    (void)in_sizes; (void)n_in; (void)out_size;

    const float* query = (const float*)d_in[0];
    const float* key   = (const float*)d_in[1];
    const float* value = (const float*)d_in[2];
    const int*   kpm   = (const int*)d_in[3];
    const float* Wq = (const float*)d_in[4];
    const float* bq = (const float*)d_in[5];
    const float* Wk = (const float*)d_in[6];
    const float* bk = (const float*)d_in[7];
    const float* Wv = (const float*)d_in[8];
    const float* bv = (const float*)d_in[9];
    const float* Wo = (const float*)d_in[10];
    const float* bo = (const float*)d_in[11];

    float* out      = (float*)d_out;
    float* attn_out = out;                              // (512,16,1024)
    float* weights  = out + (size_t)TOK * EMBED;        // (16,16,512,512)

    const size_t bufElems = (size_t)TOK * EMBED;        // 8.39M floats = 32 MB
    float* qbuf = (float*)d_ws;
    float* kbuf = qbuf + bufElems;
    float* vbuf = kbuf + bufElems;

    // attn result must not alias Q/K/V (fused kernel reads them concurrently).
    const bool bigWs = ws_size >= 4 * bufElems * sizeof(float);
    float* abuf = bigWs ? (vbuf + bufElems) : attn_out; // fallback: stage in d_out

    const dim3 gGemm(EMBED / 32, TOK / 64);             // (32, 128)

    gemm_xwT_bias<<<gGemm, 256, 0, stream>>>(query, Wq, bq, qbuf);
    gemm_xwT_bias<<<gGemm, 256, 0, stream>>>(key,   Wk, bk, kbuf);
    gemm_xwT_bias<<<gGemm, 256, 0, stream>>>(value, Wv, bv, vbuf);

    attn_fused<<<dim3(NQ / 64, BATCH * HEADS), 512, 0, stream>>>(
        qbuf, kbuf, vbuf, kpm, weights, abuf);

    if (bigWs) {
        gemm_xwT_bias<<<gGemm, 256, 0, stream>>>(abuf, Wo, bo, attn_out);
    } else {
        // abuf == attn_out region: project into qbuf (Q is dead), then copy back.
        gemm_xwT_bias<<<gGemm, 256, 0, stream>>>(abuf, Wo, bo, qbuf);
        hipMemcpyAsync(attn_out, qbuf, bufElems * sizeof(float),
                       hipMemcpyDeviceToDevice, stream);
    }
}